// ScorePredictorNoEdge_5480378269727
// MI455X (gfx1250) — compile-verified
//
#include <hip/hip_runtime.h>

// ---------------------------------------------------------------------------
// ScorePredictorNoEdge for MI455X (gfx1250, wave32)
//
// Phase 1: xs = x @ W[:128], xd = x @ W[128:]  via V_WMMA_F32_16X16X4_F32
//   - one wave32 computes a 16-node tile; A = 16x4 f32 chunk of x,
//     B = 4x16 with col0 = Ws chunk, col1 = Wd chunk (cols 2..15 zero),
//     accumulated over 32 K-chunks.
//   - B columns live zero-padded in LDS so every lane issues an
//     unconditional ds_load (no saveexec churn, EXEC stays all-1s).
//   - Full tiles store results as two global_store_b128 per active lane.
// Phase 2: out[e] = xs[src[e]] + xd[dst[e]] + b   (gathers hit L2: 400KB table)
// ---------------------------------------------------------------------------

typedef __attribute__((ext_vector_type(2))) float  v2f;
typedef __attribute__((ext_vector_type(4))) float  v4f;
typedef __attribute__((ext_vector_type(8))) float  v8f;
typedef __attribute__((ext_vector_type(2))) long long v2i64;

#define DIM      128
#define BSTRIDE  132   // column stride in LDS floats: lane n -> bank 4n (mod 64)

__global__ __launch_bounds__(256)
void node_proj_wmma(const float* __restrict__ x,
                    const float* __restrict__ W,
                    float* __restrict__ xs,
                    float* __restrict__ xd,
                    int n_nodes)
{
    // Zero-padded B table: wb[n*BSTRIDE + k] = (n==0) ? Ws[k] : (n==1) ? Wd[k] : 0
    __shared__ float wb[16 * BSTRIDE];
    for (int i = threadIdx.x; i < 16 * BSTRIDE; i += 256) {
        const int col = i / BSTRIDE;
        const int k   = i - col * BSTRIDE;
        float v = 0.0f;
        if (col < 2 && k < DIM) v = W[col * DIM + k];
        wb[i] = v;
    }
    __syncthreads();

    const int lane = threadIdx.x & 31;
    const int wave = threadIdx.x >> 5;
    const int tile = blockIdx.x * 8 + wave;      // 16 nodes per tile
    const int node_base = tile * 16;

    if (node_base < n_nodes) {                   // wave-uniform guard
        const int half = lane >> 4;              // K sub-pair select (ISA A layout)
        const int n    = lane & 15;              // A: row M; B/C/D: column N

        const int arow = min(node_base + n, n_nodes - 1);
        const float* __restrict__ xrow = x + (size_t)arow * DIM;
        const float* __restrict__ wcol = wb + n * BSTRIDE;

        v8f acc = {};
        #pragma unroll 8
        for (int c = 0; c < 32; ++c) {
            const int k0 = c * 4 + half * 2;     // 8B-aligned float pair
            v2f a  = *(const v2f*)(xrow + k0);   // global_load_b64
            v2f bm = *(const v2f*)(wcol + k0);   // ds_load_b64, conflict-free
            acc = __builtin_amdgcn_wmma_f32_16x16x4_f32(
                      false, a, false, bm, (short)0, acc, false, false);
        }

        // C/D layout: VGPR i -> M = i (lanes 0-15) or M = i+8 (lanes 16-31),
        // N = lane & 15. Column 0 = xs, column 1 = xd.
        // Active lane writes 8 consecutive floats at a 32B-aligned address.
        const int mbase = node_base + half * 8;
        if (node_base + 16 <= n_nodes) {         // full tile (always, 50000%16==0)
            if (n < 2) {
                float* __restrict__ outp = (n == 0) ? xs : xd;
                v4f lo = { acc[0], acc[1], acc[2], acc[3] };
                v4f hi = { acc[4], acc[5], acc[6], acc[7] };
                *(v4f*)(outp + mbase)     = lo;  // global_store_b128
                *(v4f*)(outp + mbase + 4) = hi;  // global_store_b128
            }
        } else if (n < 2) {                      // ragged tail tile (cold path)
            float* __restrict__ outp = (n == 0) ? xs : xd;
            for (int i = 0; i < 8; ++i) {
                const int m = mbase + i;
                if (m < n_nodes) outp[m] = acc[i];
            }
        }
    }
}

__global__ __launch_bounds__(256)
void edge_score(const long long* __restrict__ src,
                const long long* __restrict__ dst,
                const float* __restrict__ xs,
                const float* __restrict__ xd,
                const float* __restrict__ bias,
                float* __restrict__ out,
                int n_edges)
{
    // 2 edges per thread: b128 index loads, b64 result store.
    const int e0 = (blockIdx.x * 256 + threadIdx.x) * 2;
    const float bv = bias[0];
    if (e0 + 1 < n_edges) {
        v2i64 s2 = *(const v2i64*)(src + e0);    // global_load_b128
        v2i64 d2 = *(const v2i64*)(dst + e0);    // global_load_b128
        v2f r;
        r.x = xs[s2.x] + xd[d2.x] + bv;
        r.y = xs[s2.y] + xd[d2.y] + bv;
        *(v2f*)(out + e0) = r;                   // global_store_b64
    } else if (e0 < n_edges) {
        out[e0] = xs[src[e0]] + xd[dst[e0]] + bv;
    }
}

extern "C" void kernel_launch(void* const* d_in, const int* in_sizes, int n_in,
                              void* d_out, int out_size, void* d_ws, size_t ws_size,
                              hipStream_t stream)
{
    const float*     x   = (const float*)d_in[0];
    const long long* src = (const long long*)d_in[1];
    const long long* dst = (const long long*)d_in[2];
    const float*     W   = (const float*)d_in[3];
    const float*     b   = (const float*)d_in[4];
    float*           out = (float*)d_out;

    const int n_nodes = in_sizes[0] / DIM;   // 50000
    const int n_edges = in_sizes[1];         // 1.6M

    float* xs = (float*)d_ws;                // n_nodes f32
    float* xd = xs + n_nodes;                // n_nodes f32 (total 400KB << ws)

    const int tiles   = (n_nodes + 15) / 16;
    const int blocks1 = (tiles + 7) / 8;     // 8 waves (tiles) per 256-thr block
    node_proj_wmma<<<blocks1, 256, 0, stream>>>(x, W, xs, xd, n_nodes);

    const int pairs   = (n_edges + 1) / 2;
    const int blocks2 = (pairs + 255) / 256;
    edge_score<<<blocks2, 256, 0, stream>>>(src, dst, xs, xd, b, out, n_edges);
}